// GcnLayers_38439957299969
// MI455X (gfx1250) — compile-verified
//
#include <hip/hip_runtime.h>
#include <hip/hip_bf16.h>
#include <math.h>

typedef float v2f __attribute__((ext_vector_type(2)));
typedef float v8f __attribute__((ext_vector_type(8)));

#define GCN_N     100000
#define GCN_E     1600000
#define GCN_F     128          // feature width (both in and hidden)
#define TPB       256

// Guaranteed-native non-returning fp32 atomic add (avoids any CAS expansion).
// CDNA5: atomics execute in L2 (device-coherent); STOREcnt-tracked, implicit
// wait-idle at s_endpgm covers completion.
__device__ __forceinline__ void atomic_add_f32_dev(float* p, float v) {
    asm volatile("global_atomic_add_f32 %0, %1, off scope:SCOPE_DEV"
                 :
                 : "v"(p), "v"(v)
                 : "memory");
}

// ---------------------------------------------------------------- degree prep
__global__ void k_deg_init(float* __restrict__ deg, int n) {
    int i = blockIdx.x * blockDim.x + threadIdx.x;
    if (i < n) deg[i] = 1.0f;                       // self-loop contributes 1
}

__global__ void k_deg_edges(const int* __restrict__ dst, float* __restrict__ deg, int ne) {
    int e = blockIdx.x * blockDim.x + threadIdx.x;
    if (e < ne) atomic_add_f32_dev(&deg[dst[e]], 1.0f);
}

__global__ void k_deg_to_dinv(float* __restrict__ deg, int n) {
    int i = blockIdx.x * blockDim.x + threadIdx.x;
    if (i < n) deg[i] = rsqrtf(deg[i]);             // deg >= 1 always
}

// ---------------------------------------------------------------- fp32 WMMA GEMM
// C[N x 128] = A[N x 128] * W[128 x 128], row-major, exact fp32 via
// V_WMMA_F32_16X16X4_F32.  One wave -> one 16x16 tile; 8 waves/block cover a
// 16-row x 128-col stripe.  N is a multiple of 16 so EXEC is all-ones.
__global__ void __launch_bounds__(TPB) k_gemm_wmma(const float* __restrict__ A,
                                                   const float* __restrict__ W,
                                                   float* __restrict__ C) {
    const int lane = threadIdx.x & 31;
    const int wave = threadIdx.x >> 5;
    const int m    = lane & 15;     // row within tile (A) / col within tile (B)
    const int half = lane >> 4;     // 0: K={0,1}; 1: K={2,3}
    const int row0 = blockIdx.x << 4;
    const int col0 = wave << 4;

    const float* __restrict__ arow = A + (size_t)(row0 + m) * GCN_F;
    const float* __restrict__ bcol = W + col0 + m;

    v8f acc = {};
#pragma unroll
    for (int k0 = 0; k0 < GCN_F; k0 += 4) {
        const int ka = k0 + half * 2;
        v2f a = { arow[ka], arow[ka + 1] };                       // contiguous b64
        v2f b = { bcol[(size_t)ka * GCN_F],                       // coalesced across lanes
                  bcol[(size_t)(ka + 1) * GCN_F] };
        acc = __builtin_amdgcn_wmma_f32_16x16x4_f32(
                  false, a, false, b, (short)0, acc, false, false);
    }

#pragma unroll
    for (int i = 0; i < 8; ++i)
        C[(size_t)(row0 + half * 8 + i) * GCN_F + col0 + m] = acc[i];
}

// ---------------------------------------------------------------- self-loop init
// acc[i,:] = h[i,:] * dinv[i]^2   (fully overwrites acc -> deterministic replay)
__global__ void k_self_init(const float* __restrict__ h,
                            const float* __restrict__ dinv,
                            float* __restrict__ acc, long long total) {
    long long t = (long long)blockIdx.x * blockDim.x + threadIdx.x;
    if (t >= total) return;
    int node = (int)(t >> 7);
    float di = dinv[node];
    acc[t] = h[t] * di * di;
}

// ---------------------------------------------------------------- edge scatter
// One wave (32 lanes) per edge; each lane moves 4 features (b128 gather + 4
// native global_atomic_add_f32).  Memory-bound: roofline-dominant kernel
// (~5 GB of traffic over 3 layers -> ~220 us at 23.3 TB/s).
__global__ void __launch_bounds__(TPB) k_edge_scatter(const int* __restrict__ src,
                                                      const int* __restrict__ dst,
                                                      const float* __restrict__ dinv,
                                                      const float* __restrict__ h,
                                                      float* __restrict__ acc, int ne) {
    long long tid = (long long)blockIdx.x * blockDim.x + threadIdx.x;
    int e    = (int)(tid >> 5);
    int lane = (int)(tid & 31);
    if (e >= ne) return;
    int s = src[e], d = dst[e];           // wave-uniform addresses -> broadcast fill
    float norm = dinv[s] * dinv[d];
    float4 hv = *(const float4*)(h + (size_t)s * GCN_F + lane * 4);
    float* out = acc + (size_t)d * GCN_F + lane * 4;
    atomic_add_f32_dev(out + 0, hv.x * norm);
    atomic_add_f32_dev(out + 1, hv.y * norm);
    atomic_add_f32_dev(out + 2, hv.z * norm);
    atomic_add_f32_dev(out + 3, hv.w * norm);
}

// ---------------------------------------------------------------- bias(+res)+ELU
// Safe for out==acc and out==res (pure elementwise same-index).
__global__ void k_finalize(const float* __restrict__ acc,
                           const float* __restrict__ bias,
                           const float* __restrict__ res,
                           float* __restrict__ out, long long total) {
    long long t = (long long)blockIdx.x * blockDim.x + threadIdx.x;
    if (t >= total) return;
    int f = (int)(t & (GCN_F - 1));
    float v = acc[t] + bias[f];
    if (res) v += res[t];
    out[t] = v > 0.0f ? v : expm1f(v);
}

// ---------------------------------------------------------------- launcher
extern "C" void kernel_launch(void* const* d_in, const int* in_sizes, int n_in,
                              void* d_out, int out_size, void* d_ws, size_t ws_size,
                              hipStream_t stream) {
    const float* x   = (const float*)d_in[0];
    const int*   ei  = (const int*)d_in[1];         // [2, E] int32
    const float* W0  = (const float*)d_in[2];
    const float* b0  = (const float*)d_in[3];
    const float* W1  = (const float*)d_in[4];
    const float* b1  = (const float*)d_in[5];
    const float* W2  = (const float*)d_in[6];
    const float* b2  = (const float*)d_in[7];
    const int* src = ei;
    const int* dst = ei + GCN_E;

    float* wsf  = (float*)d_ws;
    float* dinv = wsf;                                        // N floats
    size_t dinv_pad = (size_t)((GCN_N + 63) & ~63);           // 256B align
    float* B0 = wsf + dinv_pad;                               // N*128 floats
    float* B1 = B0 + (size_t)GCN_N * GCN_F;                   // N*128 floats
    float* OUT = (float*)d_out;                               // N*128 floats

    const long long total = (long long)GCN_N * GCN_F;         // 12.8M
    const int gN   = (GCN_N + TPB - 1) / TPB;                 // node-wise
    const int gE   = (GCN_E + TPB - 1) / TPB;                 // edge-wise
    const int gEl  = (int)((total + TPB - 1) / TPB);          // elementwise (50000)
    const int gSc  = (int)(((long long)GCN_E * 32 + TPB - 1) / TPB); // 200000
    const int gGm  = GCN_N / 16;                              // 6250

    // degree -> dinv (recomputed every call: deterministic)
    k_deg_init   <<<gN, TPB, 0, stream>>>(dinv, GCN_N);
    k_deg_edges  <<<gE, TPB, 0, stream>>>(dst, dinv, GCN_E);
    k_deg_to_dinv<<<gN, TPB, 0, stream>>>(dinv, GCN_N);

    // ---- layer 0: cur = elu(agg(x@W0) + b0)            -> B1
    k_gemm_wmma   <<<gGm, TPB, 0, stream>>>(x, W0, B0);
    k_self_init   <<<gEl, TPB, 0, stream>>>(B0, dinv, B1, total);
    k_edge_scatter<<<gSc, TPB, 0, stream>>>(src, dst, dinv, B0, B1, GCN_E);
    k_finalize    <<<gEl, TPB, 0, stream>>>(B1, b0, nullptr, B1, total);

    // ---- layer 1: cur = elu(agg(B1@W1) + b1 + B1)      -> OUT (d_out as acc)
    k_gemm_wmma   <<<gGm, TPB, 0, stream>>>(B1, W1, B0);
    k_self_init   <<<gEl, TPB, 0, stream>>>(B0, dinv, OUT, total);
    k_edge_scatter<<<gSc, TPB, 0, stream>>>(src, dst, dinv, B0, OUT, GCN_E);
    k_finalize    <<<gEl, TPB, 0, stream>>>(OUT, b1, B1, OUT, total);

    // ---- layer 2: out = elu(agg(OUT@W2) + b2 + OUT)    -> OUT (final)
    k_gemm_wmma   <<<gGm, TPB, 0, stream>>>(OUT, W2, B0);
    k_self_init   <<<gEl, TPB, 0, stream>>>(B0, dinv, B1, total);
    k_edge_scatter<<<gSc, TPB, 0, stream>>>(src, dst, dinv, B0, B1, GCN_E);
    k_finalize    <<<gEl, TPB, 0, stream>>>(B1, b2, OUT, OUT, total);
}